// FlexibleGraphSAGE_4028679324281
// MI455X (gfx1250) — compile-verified
//
#include <hip/hip_runtime.h>

typedef __attribute__((ext_vector_type(2))) float v2f;
typedef __attribute__((ext_vector_type(8))) float v8f;

#define NN 50000
#define NE 800000
#define DH 128
#define LDP 132   // padded LDS row stride (floats): 132 % 64 = 4 -> conflict-free row reads

// ---------------- utility kernels ----------------

__global__ void zero_f32(float* __restrict__ p, int n) {
    int i = blockIdx.x * blockDim.x + threadIdx.x;
    if (i < n) p[i] = 0.0f;
}

__global__ void degree_kernel(const long long* __restrict__ dst,
                              float* __restrict__ deg, int e) {
    int i = blockIdx.x * blockDim.x + threadIdx.x;
    if (i < e) unsafeAtomicAdd(&deg[(int)dst[i]], 1.0f);
}

__global__ void invdeg_kernel(float* __restrict__ deg, int n) {
    int i = blockIdx.x * blockDim.x + threadIdx.x;
    if (i < n) deg[i] = 1.0f / fmaxf(deg[i], 1.0f);
}

// One wave (32 lanes) per edge; each lane moves 4 floats of the 128-dim row.
// h[src] is gathered (L2-resident, 25.6 MB) and atomically added into agg[dst].
__global__ void scatter_kernel(const long long* __restrict__ src,
                               const long long* __restrict__ dst,
                               const float* __restrict__ h,
                               float* __restrict__ agg, int e) {
    int t = blockIdx.x * blockDim.x + threadIdx.x;
    int eid = t >> 5;
    int lane = t & 31;
    if (eid >= e) return;
    int s = (int)src[eid];
    int d = (int)dst[eid];
    const float4 v = *(const float4*)(h + (size_t)s * DH + lane * 4);
    float* a = agg + (size_t)d * DH + lane * 4;
    unsafeAtomicAdd(a + 0, v.x);
    unsafeAtomicAdd(a + 1, v.y);
    unsafeAtomicAdd(a + 2, v.z);
    unsafeAtomicAdd(a + 3, v.w);
}

// ---------------- fused SAGE GEMM: out = relu?(agg*invdeg @ Wl^T + bl + h @ Wr^T) -------------
// Block = numColTiles waves; all waves share one 16-node tile staged in LDS
// (inv_deg folded in during staging), each wave owns one 16-column tile of the
// output and runs the K loop with V_WMMA_F32_16X16X4_F32 fed from LDS.
// blockDim.x = 32 * (dout/16); gridDim.x = NN/16.
template <bool RELU>
__global__ void sage_gemm(const float* __restrict__ agg,
                          const float* __restrict__ h,
                          const float* __restrict__ Wl,   // [dout, 128] row-major
                          const float* __restrict__ bl,   // [dout]
                          const float* __restrict__ Wr,   // [dout, 128] row-major
                          const float* __restrict__ invdeg,
                          float* __restrict__ out, int dout) {
    __shared__ float sAgg[16][LDP];   // agg tile, pre-scaled by inv_deg
    __shared__ float sH[16][LDP];     // h tile

    const int rowBase = blockIdx.x * 16;   // node tile

    // ---- stage 16x128 A tiles into LDS (coalesced float4), fold inv_deg ----
    for (int idx = threadIdx.x; idx < 16 * (DH / 4); idx += blockDim.x) {
        const int r = idx >> 5;            // 0..15
        const int c = (idx & 31) << 2;     // 0,4,..,124
        const size_t g = (size_t)(rowBase + r) * DH + c;
        const float id = invdeg[rowBase + r];
        const float4 va = *(const float4*)(agg + g);
        const float4 vh = *(const float4*)(h + g);
        sAgg[r][c + 0] = va.x * id;
        sAgg[r][c + 1] = va.y * id;
        sAgg[r][c + 2] = va.z * id;
        sAgg[r][c + 3] = va.w * id;
        *(float4*)&sH[r][c] = vh;
    }
    __syncthreads();

    const int wave = threadIdx.x >> 5;     // column tile within block
    const int lane = threadIdx.x & 31;
    const int half = lane >> 4;            // 0: lanes 0-15, 1: lanes 16-31
    const int lid  = lane & 15;

    const int col = wave * 16 + lid;       // B-operand column / C-D column (N = lid)
    const float* __restrict__ wlr = Wl + (size_t)col * DH;
    const float* __restrict__ wrr = Wr + (size_t)col * DH;

    v8f c = {};
#pragma unroll 4
    for (int k0 = 0; k0 < DH; k0 += 4) {
        // A 16x4 f32 layout: VGPR0/1 hold K={0,1} (lanes 0-15) or K={2,3} (lanes 16-31)
        const int ko = k0 + 2 * half;
        const v2f a0 = *(const v2f*)(&sAgg[lid][ko]);   // ds_load_b64, bank-conflict free
        const v2f b0 = *(const v2f*)(wlr + ko);         // B[k][col] = Wl[col][k]
        c = __builtin_amdgcn_wmma_f32_16x16x4_f32(false, a0, false, b0,
                                                  (short)0, c, false, false);
        const v2f a1 = *(const v2f*)(&sH[lid][ko]);
        const v2f b1 = *(const v2f*)(wrr + ko);         // B[k][col] = Wr[col][k]
        c = __builtin_amdgcn_wmma_f32_16x16x4_f32(false, a1, false, b1,
                                                  (short)0, c, false, false);
    }

    const float bias = bl[col];
#pragma unroll
    for (int i = 0; i < 8; ++i) {
        // C/D layout: VGPR i -> row M = i + 8*half, lane -> column N
        float v = c[i] + bias;
        if (RELU) v = fmaxf(v, 0.0f);
        out[(size_t)(rowBase + i + 8 * half) * dout + col] = v;
    }
}

// ---------------- launcher ----------------

extern "C" void kernel_launch(void* const* d_in, const int* in_sizes, int n_in,
                              void* d_out, int out_size, void* d_ws, size_t ws_size,
                              hipStream_t stream) {
    const float*     x   = (const float*)d_in[0];
    const long long* ei  = (const long long*)d_in[1];
    const float* Wl0 = (const float*)d_in[2];
    const float* bl0 = (const float*)d_in[3];
    const float* Wr0 = (const float*)d_in[4];
    const float* Wl1 = (const float*)d_in[5];
    const float* bl1 = (const float*)d_in[6];
    const float* Wr1 = (const float*)d_in[7];
    const float* Wl2 = (const float*)d_in[8];
    const float* bl2 = (const float*)d_in[9];
    const float* Wr2 = (const float*)d_in[10];
    float* out = (float*)d_out;

    const long long* src = ei;        // edge_index[0]
    const long long* dst = ei + NE;   // edge_index[1]

    // workspace layout (floats): invdeg[NN] | agg[NN*128] | h0[NN*128] | h1[NN*128]
    float* ws     = (float*)d_ws;
    float* invdeg = ws;
    float* agg    = ws + ((NN + 255) & ~255);
    float* h0     = agg + (size_t)NN * DH;
    float* h1     = h0  + (size_t)NN * DH;

    const int TB = 256;
    const int featN = NN * DH;                       // 6.4M
    const int scatThreads = NE * 32;                 // one wave per edge

    // degrees -> inverse degrees
    zero_f32<<<(NN + TB - 1) / TB, TB, 0, stream>>>(invdeg, NN);
    degree_kernel<<<(NE + TB - 1) / TB, TB, 0, stream>>>(dst, invdeg, NE);
    invdeg_kernel<<<(NN + TB - 1) / TB, TB, 0, stream>>>(invdeg, NN);

    // ---- layer 0: x[*,128] -> h0[*,128], ReLU ----
    zero_f32<<<(featN + TB - 1) / TB, TB, 0, stream>>>(agg, featN);
    scatter_kernel<<<(scatThreads + TB - 1) / TB, TB, 0, stream>>>(src, dst, x, agg, NE);
    sage_gemm<true><<<NN / 16, 8 * 32, 0, stream>>>(agg, x, Wl0, bl0, Wr0, invdeg, h0, 128);

    // ---- layer 1: h0 -> h1, ReLU ----
    zero_f32<<<(featN + TB - 1) / TB, TB, 0, stream>>>(agg, featN);
    scatter_kernel<<<(scatThreads + TB - 1) / TB, TB, 0, stream>>>(src, dst, h0, agg, NE);
    sage_gemm<true><<<NN / 16, 8 * 32, 0, stream>>>(agg, h0, Wl1, bl1, Wr1, invdeg, h1, 128);

    // ---- layer 2: h1 -> out[*,64], no ReLU ----
    zero_f32<<<(featN + TB - 1) / TB, TB, 0, stream>>>(agg, featN);
    scatter_kernel<<<(scatThreads + TB - 1) / TB, TB, 0, stream>>>(src, dst, h1, agg, NE);
    sage_gemm<false><<<NN / 16, 4 * 32, 0, stream>>>(agg, h1, Wl2, bl2, Wr2, invdeg, out, 64);
}